// MultiViewNetwork_61469571940656
// MI455X (gfx1250) — compile-verified
//
#include <hip/hip_runtime.h>
#include <math.h>

// ---------------------------------------------------------------------------
// MultiViewNetwork BEV<->spherical re-projection, MI455X (gfx1250).
// Bandwidth-bound double-scatter; no matmul -> no WMMA. CDNA5 features used:
//   * GLOBAL_LOAD_ASYNC_TO_LDS_B128 + s_wait_asynccnt (ASYNCcnt DMA path)
//   * channel-innermost 128B-blocked intermediate so scatters are b128-wide
//   * rv (64 MiB) and out (128 MiB) sized to live in the 192 MB L2
// ---------------------------------------------------------------------------

namespace {
constexpr int   B  = 16, C = 32, Hb = 256, Wb = 256, Hr = 32, Wr = 1024;
constexpr int   HWb = Hb * Wb;   // 65536
constexpr int   HWr = Hr * Wr;   // 32768
constexpr float NEG_FILL = -1e9f;
constexpr float PI_F     = 3.14159265358979323846f;
constexpr float TH_MAX_F = 0.05235987755982988f;    // radians(+3 deg)
constexpr float TH_MIN_F = -0.43633231299858238f;   // radians(-25 deg)
constexpr float TH_SPAN  = TH_MAX_F - TH_MIN_F;
constexpr float J_SCALE  = (float)(Wr - 1) / (2.0f * PI_F);
constexpr float I_SCALE  = (float)(Hr - 1) / TH_SPAN;
}

// ---- K1: fill rv (transposed intermediate) and out with NEG_FILL ----------
__global__ void __launch_bounds__(256) mvn_fill_kernel(float* __restrict__ rv,
                                                       float* __restrict__ out,
                                                       int n_rv4, int n_out4) {
  const float4 f = make_float4(NEG_FILL, NEG_FILL, NEG_FILL, NEG_FILL);
  const int stride = gridDim.x * blockDim.x;
  float4* rv4  = reinterpret_cast<float4*>(rv);
  float4* out4 = reinterpret_cast<float4*>(out);
  for (int i = blockIdx.x * blockDim.x + threadIdx.x; i < n_rv4;  i += stride) rv4[i]  = f;
  for (int i = blockIdx.x * blockDim.x + threadIdx.x; i < n_out4; i += stride) out4[i] = f;
}

// ---- K2: des_2_sph scatter ------------------------------------------------
// One block = one (b, h) row of 256 pixels. Async-DMA the 32x256 channel tile
// into LDS while computing spherical indices, then emit one contiguous 128 B
// block per pixel into rv_t[b][idx][c].
__global__ void __launch_bounds__(256) mvn_scatter1_kernel(const float* __restrict__ bev,
                                                           const float* __restrict__ zmap,
                                                           float* __restrict__ rv) {
  __shared__ float tile[C][Wb];                 // 32 KiB of 320 KiB/WGP
  const int b   = blockIdx.x >> 8;
  const int h   = blockIdx.x & 255;
  const int tid = threadIdx.x;

  // Uniform 64-bit base for this (b, h) row; per-lane 32-bit offsets.
  const float* gbase = bev + ((size_t)b * C * HWb + (size_t)h * Wb);
  const unsigned lds_base = (unsigned)(uintptr_t)(&tile[0][0]);

  // 2048 chunks of 16 B; 8 async b128 DMAs per thread, fully coalesced
  // (each wave covers 512 contiguous bytes of one channel row per issue).
#pragma unroll
  for (int k = 0; k < 8; ++k) {
    const int q        = k * 256 + tid;                       // chunk id
    const int c        = q >> 6;                              // channel
    const unsigned voff  = (unsigned)(c * (HWb * 4) + ((q & 63) << 4));
    const unsigned laddr = lds_base + (unsigned)(q << 4);
    asm volatile("global_load_async_to_lds_b128 %0, %1, %2"
                 :: "v"(laddr), "v"(voff), "s"(gbase)
                 : "memory");
  }

  // Overlap index math (VALU/trans) with the DMA.
  const int w = tid;
  const size_t zb = (size_t)b * 3 * HWb + (size_t)h * Wb + w;
  const float z = (zmap[zb] + zmap[zb + HWb] + zmap[zb + 2 * HWb]) * (1.0f / 3.0f);
  const float x = -50.0f + ((float)w + 0.5f) * (100.0f / 256.0f);
  const float y =  50.0f - ((float)h + 0.5f) * (100.0f / 256.0f);
  const float r = sqrtf(x * x + y * y + z * z) + 1e-6f;
  const float phi   = atan2f(y, x);
  const float theta = asinf(z / r);
  int j = (int)rintf((phi + PI_F) * J_SCALE);
  int i = (int)rintf((TH_MAX_F - theta) * I_SCALE);
  j = min(max(j, 0), Wr - 1);
  i = min(max(i, 0), Hr - 1);
  const int idx = i * Wr + j;

  // Drain this wave's async DMAs, then block-wide barrier.
  asm volatile("s_wait_asynccnt 0" ::: "memory");
  __syncthreads();

  // Gather this pixel's 32 channels from LDS (bank-conflict-free: address
  // stride across lanes is 4 B) and store one 128 B block (8 x b128).
  float4* dst = reinterpret_cast<float4*>(rv + ((size_t)b * HWr + (size_t)idx) * C);
#pragma unroll
  for (int k = 0; k < 8; ++k) {
    float4 v4;
    v4.x = tile[4 * k + 0][tid];
    v4.y = tile[4 * k + 1][tid];
    v4.z = tile[4 * k + 2][tid];
    v4.w = tile[4 * k + 3][tid];
    dst[k] = v4;
  }
}

// ---- K3: sph_2_des scatter ------------------------------------------------
// One thread per RV pixel: contiguous 128 B gather from rv_t (L2-resident),
// scattered per-channel stores into out (absorbed by 192 MB L2).
__global__ void __launch_bounds__(256) mvn_scatter2_kernel(const float* __restrict__ rv,
                                                           const float* __restrict__ rmap,
                                                           float* __restrict__ out) {
  const int t = blockIdx.x * 256 + threadIdx.x;   // 0 .. B*HWr-1
  const int b = t >> 15;
  const int p = t & (HWr - 1);
  const int i = p >> 10;
  const int j = p & (Wr - 1);

  const size_t rb = (size_t)b * 3 * HWr + p;
  const float r = (rmap[rb] + rmap[rb + HWr] + rmap[rb + 2 * HWr]) * (1.0f / 3.0f);
  const float phi   = -PI_F + ((float)j + 0.5f) * (2.0f * PI_F / (float)Wr);
  const float theta = TH_MAX_F - ((float)i + 0.5f) * (TH_SPAN / (float)Hr);
  const float ct = cosf(theta);
  const float x = r * ct * cosf(phi);
  const float y = r * ct * sinf(phi);
  int u = (int)rintf((x + 50.0f) * (255.0f / 100.0f));
  int v = (int)rintf((50.0f - y) * (255.0f / 100.0f));
  u = min(max(u, 0), Wb - 1);
  v = min(max(v, 0), Hb - 1);
  const int idx2 = v * Wb + u;

  const float4* src = reinterpret_cast<const float4*>(rv + ((size_t)b * HWr + p) * C);
  float* ob = out + (size_t)b * C * HWb + idx2;
#pragma unroll
  for (int k = 0; k < 8; ++k) {
    const float4 v4 = src[k];
    ob[(4 * k + 0) * HWb] = v4.x;
    ob[(4 * k + 1) * HWb] = v4.y;
    ob[(4 * k + 2) * HWb] = v4.z;
    ob[(4 * k + 3) * HWb] = v4.w;
  }
}

// ---------------------------------------------------------------------------
extern "C" void kernel_launch(void* const* d_in, const int* in_sizes, int n_in,
                              void* d_out, int out_size, void* d_ws, size_t ws_size,
                              hipStream_t stream) {
  const float* bev  = (const float*)d_in[0];   // (16,32,256,256)
  const float* zmap = (const float*)d_in[1];   // (16,3,256,256)
  const float* rmap = (const float*)d_in[2];   // (16,3,32,1024)
  float* out = (float*)d_out;                  // (16,32,256,256)
  float* rv  = (float*)d_ws;                   // rv_t[b][32768][32]: 64 MiB

  const int n_rv4  = (B * HWr * C) / 4;        // 4,194,304 float4
  const int n_out4 = (B * C * HWb) / 4;        // 8,388,608 float4

  mvn_fill_kernel<<<dim3(4096), dim3(256), 0, stream>>>(rv, out, n_rv4, n_out4);
  mvn_scatter1_kernel<<<dim3(B * Hb), dim3(256), 0, stream>>>(bev, zmap, rv);
  mvn_scatter2_kernel<<<dim3((B * HWr) / 256), dim3(256), 0, stream>>>(rv, rmap, out);
}